// LightGCN_62457414419242
// MI455X (gfx1250) — compile-verified
//
#include <hip/hip_runtime.h>
#include <hip/hip_bf16.h>
#include <math.h>

typedef float v2f __attribute__((ext_vector_type(2)));
typedef float v8f __attribute__((ext_vector_type(8)));

#define EMB_DIM 64

// ---------------------------------------------------------------------------
// Build E = concat(user_emb, item_emb) into both acc and cur.
// ---------------------------------------------------------------------------
__global__ void lgcn_init(const float* __restrict__ ue, const float* __restrict__ ie,
                          float* __restrict__ acc, float* __restrict__ cur,
                          long long nU64, long long nd) {
    long long i = (long long)blockIdx.x * blockDim.x + threadIdx.x;
    if (i >= nd) return;
    float v = (i < nU64) ? ue[i] : ie[i - nU64];
    acc[i] = v;
    cur[i] = v;
}

// ---------------------------------------------------------------------------
// One SpMM layer: nxt[row] += val * cur[col], 16 lanes x float4 per edge.
// cur/nxt are 38.4 MB each -> both resident in the 192 MB L2, so the random
// gather/scatter stays on-chip. Prefetch the streaming adjacency ahead.
// ---------------------------------------------------------------------------
__global__ void lgcn_spmm(const int* __restrict__ rows, const int* __restrict__ cols,
                          const float* __restrict__ vals,
                          const float* __restrict__ cur, float* __restrict__ nxt,
                          long long nquads, long long nnz2) {
    long long stride = (long long)gridDim.x * blockDim.x;
    for (long long i = (long long)blockIdx.x * blockDim.x + threadIdx.x;
         i < nquads; i += stride) {
        long long e = i >> 4;
        int q = (int)(i & 15);
        int r = rows[e];
        int c = cols[e];
        float v = vals[e];
        if (e + 2048 < nnz2) {
            __builtin_prefetch(cols + e + 2048, 0, 0);   // global_prefetch_b8
            __builtin_prefetch(rows + e + 2048, 0, 0);
        }
        const float4 x = *reinterpret_cast<const float4*>(cur + (long long)c * EMB_DIM + q * 4);
        float* dst = nxt + (long long)r * EMB_DIM + q * 4;
        atomicAdd(dst + 0, v * x.x);
        atomicAdd(dst + 1, v * x.y);
        atomicAdd(dst + 2, v * x.z);
        atomicAdd(dst + 3, v * x.w);
    }
}

// ---------------------------------------------------------------------------
// acc += nxt
// ---------------------------------------------------------------------------
__global__ void lgcn_axpy(float* __restrict__ acc, const float* __restrict__ nxt, long long nd) {
    long long i = (long long)blockIdx.x * blockDim.x + threadIdx.x;
    if (i < nd) acc[i] += nxt[i];
}

// ---------------------------------------------------------------------------
// Component select from a WMMA f32 accumulator fragment.
// ---------------------------------------------------------------------------
__device__ __forceinline__ float pick8(v8f c, int i) {
    switch (i) {
        case 0: return c[0];
        case 1: return c[1];
        case 2: return c[2];
        case 3: return c[3];
        case 4: return c[4];
        case 5: return c[5];
        case 6: return c[6];
        default: return c[7];
    }
}

// ---------------------------------------------------------------------------
// BPR loss + reg. One wave (32 lanes) per 16-row batch tile.
// Gram blocks S_pos = U * P^T and S_neg = U * N^T computed with
// V_WMMA_F32_16X16X4_F32 over 16 K-steps (K=64 total), diagonal extracted.
//
// Fragment layouts (CDNA5 ISA 7.12.2, f32):
//   A 16x4 : lane L holds A[L&15][kb + 2*(L>>4) + {0,1}] in {a.x, a.y}
//   B 4x16 : lane L holds B[2*(L>>4)+{0,1}][L&15]  == rowT[L&15][kb+2*(L>>4)+{0,1}]
//   C 16x16: lanes 0-15: N=lane, VGPR r -> M=r ; lanes 16-31: N=lane-16, M=r+8
//   => diag m=0..7 in lane m comp m ; diag m=8..15 in lane m+16 comp m-8
// ---------------------------------------------------------------------------
__global__ void lgcn_loss(const int* __restrict__ user_ids, const int* __restrict__ pos_ids,
                          const int* __restrict__ neg_ids,
                          const float* __restrict__ Esum,     // sum over layers (pre /4)
                          const float* __restrict__ ue, const float* __restrict__ ie,
                          long long n_users,
                          float* __restrict__ lossAcc, float* __restrict__ regAcc) {
    const int tile = blockIdx.x;
    const int lane = threadIdx.x;        // 0..31, one wave
    const int m    = lane & 15;
    const int hi   = lane >> 4;

    const int bi  = tile * 16 + m;
    const int uid = user_ids[bi];
    const int pid = pos_ids[bi];
    const int nid = neg_ids[bi];

    const float* urow = Esum + (long long)uid * EMB_DIM;
    const float* prow = Esum + (n_users + (long long)pid) * EMB_DIM;
    const float* nrow = Esum + (n_users + (long long)nid) * EMB_DIM;

    v8f cp = {0.f, 0.f, 0.f, 0.f, 0.f, 0.f, 0.f, 0.f};
    v8f cn = {0.f, 0.f, 0.f, 0.f, 0.f, 0.f, 0.f, 0.f};

#pragma unroll
    for (int kb = 0; kb < EMB_DIM; kb += 4) {
        const int col = kb + 2 * hi;
        v2f a  = {urow[col], urow[col + 1]};
        v2f bp = {prow[col], prow[col + 1]};
        v2f bn = {nrow[col], nrow[col + 1]};
        // D = A x B + C, full fp32
        cp = __builtin_amdgcn_wmma_f32_16x16x4_f32(false, a, false, bp, (short)0, cp, false, false);
        cn = __builtin_amdgcn_wmma_f32_16x16x4_f32(false, a, false, bn, (short)0, cn, false, false);
    }

    // Diagonal extraction. Both operands carry un-divided layer sums; E_final
    // = acc/4, so scores scale by 1/16.
    const bool valid = (lane < 8) || (lane >= 24);
    const int  comp  = (lane < 8) ? lane : (lane - 24);
    const float dp = pick8(cp, comp) * (1.0f / 16.0f);
    const float dn = pick8(cn, comp) * (1.0f / 16.0f);

    if (valid) {
        const float x = dp - dn;
        // -log_sigmoid(x) = softplus(-x), numerically stable
        const float t = (x >= 0.0f) ? log1pf(expf(-x)) : (-x + log1pf(expf(x)));
        atomicAdd(lossAcc, t);
    }

    // L2 reg on RAW embeddings: each lane handles 2 columns across all 16 rows.
    float rsum = 0.0f;
    const int c0 = 2 * lane;
#pragma unroll 4
    for (int r = 0; r < 16; ++r) {
        const int rb = tile * 16 + r;
        const float* a = ue + (long long)user_ids[rb] * EMB_DIM + c0;
        const float* b = ie + (long long)pos_ids[rb]  * EMB_DIM + c0;
        const float* c = ie + (long long)neg_ids[rb]  * EMB_DIM + c0;
        rsum += a[0] * a[0] + a[1] * a[1];
        rsum += b[0] * b[0] + b[1] * b[1];
        rsum += c[0] * c[0] + c[1] * c[1];
    }
    for (int off = 16; off; off >>= 1) rsum += __shfl_xor(rsum, off, 32);
    if (lane == 0) atomicAdd(regAcc, rsum);
}

// ---------------------------------------------------------------------------
// out = loss/B + 1e-4 * reg/B
// ---------------------------------------------------------------------------
__global__ void lgcn_final(const float* __restrict__ accum, float* __restrict__ out, float invB) {
    out[0] = accum[0] * invB + 1e-4f * accum[1] * invB;
}

extern "C" void kernel_launch(void* const* d_in, const int* in_sizes, int n_in,
                              void* d_out, int out_size, void* d_ws, size_t ws_size,
                              hipStream_t stream) {
    const int*   user_ids = (const int*)d_in[0];
    const int*   pos_ids  = (const int*)d_in[1];
    const int*   neg_ids  = (const int*)d_in[2];
    const float* user_emb = (const float*)d_in[3];
    const float* item_emb = (const float*)d_in[4];
    const int*   adj_rows = (const int*)d_in[5];
    const int*   adj_cols = (const int*)d_in[6];
    const float* adj_vals = (const float*)d_in[7];

    const int       batch = in_sizes[0];
    const long long nU    = in_sizes[3] / EMB_DIM;
    const long long nI    = in_sizes[4] / EMB_DIM;
    const long long ND    = (nU + nI) * (long long)EMB_DIM;
    const long long nnz2  = in_sizes[5];

    float* acc   = (float*)d_ws;            // layer-sum accumulator
    float* bufB  = acc + ND;                 // ping
    float* bufC  = bufB + ND;                // pong
    float* accum = bufC + ND;                // [0]=loss, [1]=reg

    hipMemsetAsync(accum, 0, 2 * sizeof(float), stream);

    const int tb = 256;
    const long long ndBlocks = (ND + tb - 1) / tb;
    lgcn_init<<<dim3((unsigned)ndBlocks), tb, 0, stream>>>(user_emb, item_emb, acc, bufB,
                                                           nU * EMB_DIM, ND);

    float* cur = bufB;
    float* nxt = bufC;
    const long long nquads = nnz2 * 16;
    long long spmmBlocks = (nquads + tb - 1) / tb;
    if (spmmBlocks > 65536) spmmBlocks = 65536;   // grid-stride loop

    for (int layer = 0; layer < 3; ++layer) {
        hipMemsetAsync(nxt, 0, (size_t)ND * sizeof(float), stream);
        lgcn_spmm<<<dim3((unsigned)spmmBlocks), tb, 0, stream>>>(adj_rows, adj_cols, adj_vals,
                                                                 cur, nxt, nquads, nnz2);
        lgcn_axpy<<<dim3((unsigned)ndBlocks), tb, 0, stream>>>(acc, nxt, ND);
        float* t = cur; cur = nxt; nxt = t;
    }

    // 512 tiles of 16 batch rows, one wave32 each
    lgcn_loss<<<dim3(batch / 16), 32, 0, stream>>>(user_ids, pos_ids, neg_ids, acc,
                                                   user_emb, item_emb, nU,
                                                   accum, accum + 1);
    lgcn_final<<<1, 1, 0, stream>>>(accum, (float*)d_out, 1.0f / (float)batch);
}